// YOLOv2Postprocess_49306224558218
// MI455X (gfx1250) — compile-verified
//
#include <hip/hip_runtime.h>
#include <math.h>
#include <stdint.h>

#define NN 8
#define AA 5
#define HH 20
#define WW 20
#define CC 425
#define MM 2000
#define NCLS 80
#define STRIDEC 85
#define HWc 400
#define SORTN 2048
#define MROW 64            // mask row length in dwords (2000 bits padded to 2048)
#define NMS_T 0.45f
#define SCORE_T 0.5f

// ---------------------------------------------------------------------------
// Phase 1: decode. One thread per (n, a, h, w) prediction; w is the fastest
// varying dim across consecutive lanes so the 85 strided channel reads are
// 20-wide coalesced segments. fp32 throughout (exact threshold semantics).
// ---------------------------------------------------------------------------
__global__ __launch_bounds__(256) void decode_kernel(
    const float* __restrict__ in, const int* __restrict__ imgSize,
    float* __restrict__ wsBoxes, float* __restrict__ wsScores,
    int* __restrict__ wsLabels, float* __restrict__ wsSkey)
{
  int t = blockIdx.x * 256 + threadIdx.x;
  if (t >= NN * AA * HH * WW) return;
  int w = t % WW;
  int h = (t / WW) % HH;
  int a = (t / (WW * HH)) % AA;
  int n = t / (WW * HH * AA);
  const int base = (n * CC + a * STRIDEC) * HWc + h * WW + w;

  // argmax over class logits (first-max wins, matching jnp.argmax)
  float best = in[base];
  int bi = 0;
#pragma unroll 4
  for (int s = 1; s < NCLS; ++s) {
    float v = in[base + s * HWc];
    if (v > best) { best = v; bi = s; }
  }
  float tx  = in[base + 80 * HWc];
  float ty  = in[base + 81 * HWc];
  float tw  = in[base + 82 * HWc];
  float th  = in[base + 83 * HWc];
  float obj = in[base + 84 * HWc];

  float score = 1.0f / (1.0f + expf(-obj));
  float sigx  = 1.0f / (1.0f + expf(-tx));
  float sigy  = 1.0f / (1.0f + expf(-ty));

  float aw = (a == 0) ? 30.f : (a == 1) ? 70.f : (a == 2) ? 120.f : (a == 3) ? 200.f : 320.f;
  float ah = (a == 0) ? 35.f : (a == 1) ? 80.f : (a == 2) ? 130.f : (a == 3) ? 240.f : 310.f;
  float pw = aw * expf(tw);
  float ph = ah * expf(th);

  float cx = ((float)w + sigx) * 32.0f;
  float cy = ((float)h + sigy) * 32.0f;
  float img = (float)imgSize[0];
  float hx = pw * 0.5f, hy = ph * 0.5f;
  float x1 = fminf(fmaxf(cx - hx, 0.f), img);
  float y1 = fminf(fmaxf(cy - hy, 0.f), img);
  float x2 = fminf(fmaxf(cx + hx, 0.f), img);
  float y2 = fminf(fmaxf(cy + hy, 0.f), img);

  int m  = (h * WW + w) * AA + a;   // (H,W,A) flatten order of the reference
  int gi = n * MM + m;
  float4 b4 = {x1, y1, x2, y2};
  ((float4*)wsBoxes)[gi] = b4;
  wsScores[gi] = score;
  wsLabels[gi] = bi;
  wsSkey[gi]   = (score > SCORE_T) ? score : -1.0f;   // jnp.where(valid, score, -1)
}

// ---------------------------------------------------------------------------
// Phase 2: per-image stable descending bitonic argsort over 2048 LDS slots
// (key = masked score, payload = original index; ties break on lower index,
// matching stable argsort of -s_key). One 32-wave workgroup per image.
// Writes boxes_s / scores_s / labels_s to d_out and emits the class-offset
// boxes (+label*(img+1)) and areas consumed by the mask builder.
// ---------------------------------------------------------------------------
__global__ __launch_bounds__(1024) void sort_kernel(
    const float* __restrict__ wsBoxes, const float* __restrict__ wsScores,
    const int* __restrict__ wsLabels, const float* __restrict__ wsSkey,
    const int* __restrict__ imgSize,
    float* __restrict__ outBoxes, float* __restrict__ outScores,
    float* __restrict__ outLabels, float* __restrict__ wsSkeySorted,
    float4* __restrict__ wsOBox, float* __restrict__ wsOArea)
{
  __shared__ float sk[SORTN];
  __shared__ int   si[SORTN];
  const int n = blockIdx.x, tid = threadIdx.x;

  for (int j = tid; j < SORTN; j += 1024) {
    sk[j] = (j < MM) ? wsSkey[(size_t)n * MM + j] : -3.0f;  // pad sorts last
    si[j] = j;
  }
  __syncthreads();

  for (int k = 2; k <= SORTN; k <<= 1) {
    for (int st = k >> 1; st > 0; st >>= 1) {
      for (int i = tid; i < SORTN; i += 1024) {
        int ixj = i ^ st;
        if (ixj > i) {
          float ka = sk[i];  int ia = si[i];
          float kb = sk[ixj]; int ib = si[ixj];
          // "before(a,b)": a precedes b in final order (desc key, asc idx)
          bool beforeBA = (kb > ka) || ((kb == ka) && (ib < ia));
          bool beforeAB = (ka > kb) || ((ka == kb) && (ia < ib));
          bool up = ((i & k) == 0);
          if (up ? beforeBA : beforeAB) {
            sk[i] = kb; si[i] = ib;
            sk[ixj] = ka; si[ixj] = ia;
          }
        }
      }
      __syncthreads();
    }
  }

  const float img1 = (float)imgSize[0] + 1.0f;
  for (int j = tid; j < MM; j += 1024) {
    int o = si[j];
    size_t gi = (size_t)n * MM + j, go = (size_t)n * MM + o;
    float4 b = ((const float4*)wsBoxes)[go];
    ((float4*)outBoxes)[gi] = b;
    outScores[gi]    = wsScores[go];
    float off        = (float)wsLabels[go] * img1;
    outLabels[gi]    = (float)wsLabels[go];
    wsSkeySorted[gi] = sk[j];
    float4 ob = {b.x + off, b.y + off, b.z + off, b.w + off};
    wsOBox[gi]  = ob;
    wsOArea[gi] = fmaxf(b.z - b.x, 0.f) * fmaxf(b.w - b.y, 0.f);
  }
}

// ---------------------------------------------------------------------------
// Phase 3a: parallel suppression-mask build. Each block owns a 32-row stripe
// of one image's 2000x2000 IoU matrix; all 2000 offset boxes + areas are
// staged into LDS via CDNA5 async global->LDS copies (ASYNCcnt path).
// mask[i][d] bit b = ( j = 32d+b,  j > i  &&  IoU(i,j) > 0.45 ).
// Rows are 64 dwords (2048 bits); bits for j >= 2000 are zero.
// ---------------------------------------------------------------------------
__global__ __launch_bounds__(256) void mask_kernel(
    const float4* __restrict__ wsOBox, const float* __restrict__ wsOArea,
    unsigned* __restrict__ wsMask)
{
  __shared__ float4 sbox[MM];   // 32 KB
  __shared__ float  sarea[MM];  //  8 KB
  const int n = blockIdx.y, tid = threadIdx.x;
  const int tileBase = blockIdx.x * 32;

  const uint64_t gb = (uint64_t)(uintptr_t)(wsOBox + (size_t)n * MM);
  const uint64_t ga = (uint64_t)(uintptr_t)(wsOArea + (size_t)n * MM);
  for (int j = tid; j < MM; j += 256) {
    uint32_t lb = (uint32_t)(uintptr_t)(&sbox[j]);
    uint32_t la = (uint32_t)(uintptr_t)(&sarea[j]);
    int ob = j * 16, oa = j * 4;
    asm volatile("global_load_async_to_lds_b128 %0, %1, %2"
                 :: "v"(lb), "v"(ob), "s"(gb) : "memory");
    asm volatile("global_load_async_to_lds_b32 %0, %1, %2"
                 :: "v"(la), "v"(oa), "s"(ga) : "memory");
  }
  asm volatile("s_wait_asynccnt 0x0" ::: "memory");
  __syncthreads();

  // 32 rows x 64 dwords = 2048 mask words per block, 8 per thread
  for (int item = tid; item < 32 * MROW; item += 256) {
    int r = item >> 6;          // row within tile
    int d = item & 63;          // dword within row
    int i = tileBase + r;
    if (i >= MM) continue;
    float4 bi = sbox[i];
    float  ai = sarea[i];
    unsigned m = 0;
#pragma unroll 4
    for (int b = 0; b < 32; ++b) {
      int j = (d << 5) + b;
      if (j < MM && j > i) {
        float4 bj = sbox[j];
        float iw = fminf(bi.z, bj.z) - fmaxf(bi.x, bj.x);
        float ih = fminf(bi.w, bj.w) - fmaxf(bi.y, bj.y);
        float inter = fmaxf(iw, 0.f) * fmaxf(ih, 0.f);
        float iou = inter / fmaxf(ai + sarea[j] - inter, 1e-9f);
        m |= (iou > NMS_T) ? (1u << b) : 0u;
      }
    }
    wsMask[((size_t)n * MM + i) * MROW + d] = m;
  }
}

// ---------------------------------------------------------------------------
// Phase 3b: barrier-free serial reduce, ONE wave32 per image. The 2048-bit
// "removed" vector lives in 2 VGPRs per lane (lane l owns dwords l, l+32).
// Invalid boxes are pre-folded into the initial removed state, so
// final keep = !removed, exactly matching the reference greedy recurrence.
// 1-row-ahead register prefetch + global_prefetch_b8 16 rows ahead hide L2
// latency on the per-step row fetch.
// ---------------------------------------------------------------------------
__global__ __launch_bounds__(32) void nms_serial_kernel(
    const unsigned* __restrict__ wsMask, const float* __restrict__ wsSkeySorted,
    float* __restrict__ outKeep)
{
  const int n = blockIdx.x, lane = threadIdx.x;
  const unsigned* mrow = wsMask + (size_t)n * MM * MROW;
  const float* skey = wsSkeySorted + (size_t)n * MM;

  // initial removed state: invalid (or padding) boxes start removed
  unsigned rem0 = 0, rem1 = 0;
#pragma unroll 4
  for (int b = 0; b < 32; ++b) {
    int j0 = (lane << 5) + b;            // dword `lane`,   j in [0,1024)
    int j1 = ((lane + 32) << 5) + b;     // dword `lane+32`, j in [1024,2048)
    if (skey[j0] <= 0.0f) rem0 |= (1u << b);
    if (j1 >= MM || skey[j1] <= 0.0f) rem1 |= (1u << b);
  }

  // prefetch row 0
  unsigned ra = mrow[lane], rb = mrow[32 + lane];
  for (int i = 0; i < MM; ++i) {
    if (i + 16 < MM)
      __builtin_prefetch(mrow + (size_t)(i + 16) * MROW + lane, 0, 0);
    unsigned na = 0, nb = 0;
    if (i + 1 < MM) {
      na = mrow[(size_t)(i + 1) * MROW + lane];
      nb = mrow[(size_t)(i + 1) * MROW + 32 + lane];
    }
    int d = i >> 5, bit = i & 31;
    unsigned cur = (d < 32) ? (unsigned)__shfl((int)rem0, d)
                            : (unsigned)__shfl((int)rem1, d - 32);
    if (!((cur >> bit) & 1u)) {          // box i is kept -> suppress its row
      rem0 |= ra;
      rem1 |= rb;
    }
    ra = na; rb = nb;
  }

  // expand removed bits to keep floats
#pragma unroll
  for (int s = 0; s < 2; ++s) {
    unsigned r = s ? rem1 : rem0;
    int dbase = (s ? (lane + 32) : lane) << 5;
    for (int b = 0; b < 32; ++b) {
      int j = dbase + b;
      if (j < MM)
        outKeep[(size_t)n * MM + j] = ((r >> b) & 1u) ? 0.0f : 1.0f;
    }
  }
}

// ---------------------------------------------------------------------------
extern "C" void kernel_launch(void* const* d_in, const int* in_sizes, int n_in,
                              void* d_out, int out_size, void* d_ws, size_t ws_size,
                              hipStream_t stream) {
  (void)in_sizes; (void)n_in; (void)out_size; (void)ws_size;
  const float* in      = (const float*)d_in[0];
  const int*   imgSize = (const int*)d_in[1];

  float*  wsBoxes      = (float*)d_ws;                   // 64000 f
  float*  wsScores     = wsBoxes + NN * MM * 4;          // 16000 f
  int*    wsLabels     = (int*)(wsScores + NN * MM);     // 16000 i
  float*  wsSkey       = (float*)(wsLabels + NN * MM);   // 16000 f
  float*  wsSkeySorted = wsSkey + NN * MM;               // 16000 f
  float4* wsOBox       = (float4*)(wsSkeySorted + NN * MM);      // 16000 float4 (16B aligned)
  float*  wsOArea      = (float*)(wsOBox + NN * MM);             // 16000 f
  unsigned* wsMask     = (unsigned*)(wsOArea + NN * MM);         // 1,024,000 u32

  float* outBoxes  = (float*)d_out;
  float* outScores = outBoxes + NN * MM * 4;
  float* outLabels = outScores + NN * MM;
  float* outKeep   = outLabels + NN * MM;

  decode_kernel<<<(NN * AA * HH * WW + 255) / 256, 256, 0, stream>>>(
      in, imgSize, wsBoxes, wsScores, wsLabels, wsSkey);
  sort_kernel<<<NN, 1024, 0, stream>>>(
      wsBoxes, wsScores, wsLabels, wsSkey, imgSize,
      outBoxes, outScores, outLabels, wsSkeySorted, wsOBox, wsOArea);
  mask_kernel<<<dim3((MM + 31) / 32, NN), 256, 0, stream>>>(
      wsOBox, wsOArea, wsMask);
  nms_serial_kernel<<<NN, 32, 0, stream>>>(
      wsMask, wsSkeySorted, outKeep);
}